// Top2Gate_6236292514564
// MI455X (gfx1250) — compile-verified
//
#include <hip/hip_runtime.h>
#include <hip/hip_bf16.h>
#include <math.h>

typedef float v2f __attribute__((ext_vector_type(2)));
typedef float v4f __attribute__((ext_vector_type(4)));
typedef float v8f __attribute__((ext_vector_type(8)));

#define S_TOK 4096
#define D_DIM 2048
#define E_EXP 8
#define CAP   8192                      // 2*S
#define EPS_F 1.1920928955078125e-07f   // FLT_EPSILON

// ---- workspace layout (bytes) ----
#define WS_LOGITS   0
#define WS_IDX1     (WS_LOGITS + S_TOK*E_EXP*4)
#define WS_IDX2     (WS_IDX1 + S_TOK*4)
#define WS_LOC1     (WS_IDX2 + S_TOK*4)
#define WS_LOC2     (WS_LOC1 + S_TOK*4)
#define WS_G1       (WS_LOC2 + S_TOK*4)
#define WS_G2       (WS_G1 + S_TOK*4)
#define WS_GSUM     (WS_G2 + S_TOK*4)
#define WS_CNT1     (WS_GSUM + E_EXP*4)

// ---------------- init per-call accumulators ----------------
__global__ void k_init(float* __restrict__ gsum, int* __restrict__ cnt1) {
    if (threadIdx.x < E_EXP) { gsum[threadIdx.x] = 0.f; cnt1[threadIdx.x] = 0; }
}

// ---------------- logits = x @ wg^T via V_WMMA_F32_16X16X4_F32 ----------------
// One wave per 16-row tile of x. 256 waves total (32 blocks x 8 waves).
// A (16x4 f32): lanes 0-15 -> rows, VGPR0=K0 VGPR1=K1; lanes 16-31 -> K2,K3.
// B (4x16 f32): row striped across lanes: lanes 0-15 K0/K1, lanes 16-31 K2/K3, N=lane&15.
// Padding columns n=8..15 alias wg row (n&7) and are zeroed with a branch-free
// cndmask select (no exec-mask manipulation in the hot loop; WMMA needs EXEC all-1s).
__global__ __launch_bounds__(256)
void k_logits_wmma(const float* __restrict__ x, const float* __restrict__ wg,
                   float* __restrict__ logits) {
    const int lane = threadIdx.x & 31;
    const int wave = blockIdx.x * 8 + (threadIdx.x >> 5);   // 0..255
    const int row0 = wave * 16;
    const int half = lane >> 4;      // 0: K0/K1, 1: K2/K3
    const int m    = lane & 15;      // A row / B column
    const bool valid = (m < E_EXP);

    const float* xrow = x + (size_t)(row0 + m) * D_DIM + half * 2;
    const float* wrow = wg + (size_t)(m & (E_EXP - 1)) * D_DIM + half * 2;

    v8f c = {};
    for (int kk = 0; kk < D_DIM; kk += 4) {
        // streamed A tile: non-temporal b64 load
        v2f a = __builtin_nontemporal_load((const v2f*)(xrow + kk));
        // unconditional B load from aliased (always-valid) row, then select-zero
        const float bx = wrow[kk];
        const float by = wrow[kk + 1];
        v2f b;
        b.x = valid ? bx : 0.f;
        b.y = valid ? by : 0.f;
        // 8 args: (neg_a, A, neg_b, B, c_mod, C, reuse_a, reuse_b)
        c = __builtin_amdgcn_wmma_f32_16x16x4_f32(false, a, false, b,
                                                  (short)0, c, false, false);
    }
    // D: VGPR g -> row0 + g (lanes 0-15) / row0 + 8 + g (lanes 16-31), col = m
    if (valid) {
        const int rbase = row0 + (half ? 8 : 0);
#pragma unroll
        for (int g = 0; g < 8; ++g)
            logits[(size_t)(rbase + g) * E_EXP + m] = c[g];
    }
}

// ---------------- per-token softmax / top1 / gumbel top2 ----------------
__global__ __launch_bounds__(256)
void k_top2(const float* __restrict__ logits, const float* __restrict__ noise_u,
            int* __restrict__ idx1, int* __restrict__ idx2,
            float* __restrict__ g1, float* __restrict__ g2,
            float* __restrict__ gsum, int* __restrict__ cnt1) {
    __shared__ float s_gs[E_EXP];
    __shared__ int   s_c1[E_EXP];
    if (threadIdx.x < E_EXP) { s_gs[threadIdx.x] = 0.f; s_c1[threadIdx.x] = 0; }
    __syncthreads();

    const int s = blockIdx.x * blockDim.x + threadIdx.x;   // 0..4095
    float l[E_EXP], p[E_EXP];
    float mx = -INFINITY;
#pragma unroll
    for (int e = 0; e < E_EXP; ++e) {
        l[e] = logits[(size_t)s * E_EXP + e];
        mx = fmaxf(mx, l[e]);
    }
    float sum = 0.f;
#pragma unroll
    for (int e = 0; e < E_EXP; ++e) { p[e] = __expf(l[e] - mx); sum += p[e]; }
    const float inv = 1.f / sum;

    // top-1 on logits (== argmax of softmax; first-max tie-break like jnp.argmax)
    int e1 = 0; float best = l[0];
#pragma unroll
    for (int e = 1; e < E_EXP; ++e) if (l[e] > best) { best = l[e]; e1 = e; }

    // top-2 on logits + Gumbel(0,1) with top-1 masked to -inf
    int e2 = 0; float best2 = -INFINITY;
#pragma unroll
    for (int e = 0; e < E_EXP; ++e) {
        if (e == e1) continue;
        const float u  = noise_u[(size_t)s * E_EXP + e];
        const float lw = l[e] - __logf(-__logf(u));   // l + (-log(-log u))
        if (lw > best2) { best2 = lw; e2 = e; }
    }

    const float p1 = p[e1] * inv, p2 = p[e2] * inv;
    const float denom = fmaxf(p1 + p2, EPS_F);
    g1[s] = p1 / denom;
    g2[s] = p2 / denom;
    idx1[s] = e1;
    idx2[s] = e2;

    // accumulate me (per-expert mean of gates) and ce counts
#pragma unroll
    for (int e = 0; e < E_EXP; ++e) atomicAdd(&s_gs[e], p[e] * inv);
    atomicAdd(&s_c1[e1], 1);
    __syncthreads();
    if (threadIdx.x < E_EXP) {
        atomicAdd(&gsum[threadIdx.x], s_gs[threadIdx.x]);
        atomicAdd(&cnt1[threadIdx.x], s_c1[threadIdx.x]);
    }
}

// ---------------- per-expert prefix counts via wave32 ballot scan ----------------
// 16 waves: wave e (0..7) scans mask1 for expert e; wave 8+e scans mask2
// (with +cnt1[e] offset = sum(mask1) over all tokens).
__global__ __launch_bounds__(512)
void k_scan(const int* __restrict__ idx1, const int* __restrict__ idx2,
            const int* __restrict__ cnt1,
            int* __restrict__ loc1, int* __restrict__ loc2) {
    const int wave = threadIdx.x >> 5;
    const int lane = threadIdx.x & 31;
    const int e = wave & 7;
    const bool second = (wave >= 8);
    const int* idx = second ? idx2 : idx1;
    int*       loc = second ? loc2 : loc1;
    const int base = second ? cnt1[e] : 0;
    int running = 0;
    for (int it = 0; it < S_TOK; it += 32) {
        const int s = it + lane;
        const bool m = (idx[s] == e);
        const unsigned mask32 = (unsigned)__ballot(m);
        const int prefix = __popc(mask32 & ((1u << lane) - 1u));
        if (m) loc[s] = base + running + prefix;   // cumsum - 1
        running += __popc(mask32);
    }
}

// ---------------- zero-fill output (dominant cost: ~2.15 GB of stores) ----------------
__global__ void k_zero(v4f* __restrict__ out4, size_t n4,
                       float* __restrict__ out, size_t tail_start, size_t n_tail) {
    size_t i = (size_t)blockIdx.x * blockDim.x + threadIdx.x;
    const size_t stride = (size_t)gridDim.x * blockDim.x;
    const v4f z = {0.f, 0.f, 0.f, 0.f};
    for (; i < n4; i += stride)
        __builtin_nontemporal_store(z, &out4[i]);
    if (blockIdx.x == 0 && threadIdx.x < n_tail)
        out[tail_start + threadIdx.x] = 0.f;
}

// ---------------- scatter the 2 nonzeros per token + l_aux ----------------
__global__ __launch_bounds__(256)
void k_scatter(const int* __restrict__ idx1, const int* __restrict__ idx2,
               const int* __restrict__ loc1, const int* __restrict__ loc2,
               const float* __restrict__ g1, const float* __restrict__ g2,
               const float* __restrict__ gsum, const int* __restrict__ cnt1,
               float* __restrict__ out) {
    const int s = blockIdx.x * blockDim.x + threadIdx.x;
    float* combine  = out + 1;
    float* dispatch = out + 1 + (size_t)S_TOK * E_EXP * CAP;
    const float a = g1[s], b = g2[s];
    const size_t o1 = ((size_t)s * E_EXP + idx1[s]) * CAP + loc1[s];
    const size_t o2 = ((size_t)s * E_EXP + idx2[s]) * CAP + loc2[s];
    combine[o1]  = a;
    dispatch[o1] = (a != 0.f) ? 1.f : 0.f;
    combine[o2]  = b;
    dispatch[o2] = (b != 0.f) ? 1.f : 0.f;
    if (s == 0) {
        float acc = 0.f;
#pragma unroll
        for (int e = 0; e < E_EXP; ++e)
            acc += (gsum[e] / (float)S_TOK) * ((float)cnt1[e] / (float)S_TOK);
        out[0] = acc / (float)E_EXP;   // l_aux = mean(me * ce)
    }
}

extern "C" void kernel_launch(void* const* d_in, const int* in_sizes, int n_in,
                              void* d_out, int out_size, void* d_ws, size_t ws_size,
                              hipStream_t stream) {
    const float* x       = (const float*)d_in[0];
    const float* wg      = (const float*)d_in[1];
    const float* noise_u = (const float*)d_in[2];
    float* out = (float*)d_out;
    char* ws = (char*)d_ws;

    float* logits = (float*)(ws + WS_LOGITS);
    int*   idx1   = (int*)  (ws + WS_IDX1);
    int*   idx2   = (int*)  (ws + WS_IDX2);
    int*   loc1   = (int*)  (ws + WS_LOC1);
    int*   loc2   = (int*)  (ws + WS_LOC2);
    float* g1     = (float*)(ws + WS_G1);
    float* g2     = (float*)(ws + WS_G2);
    float* gsum   = (float*)(ws + WS_GSUM);
    int*   cnt1   = (int*)  (ws + WS_CNT1);

    k_init<<<1, 32, 0, stream>>>(gsum, cnt1);
    k_logits_wmma<<<S_TOK / 16 / 8, 256, 0, stream>>>(x, wg, logits);
    k_top2<<<S_TOK / 256, 256, 0, stream>>>(logits, noise_u, idx1, idx2, g1, g2, gsum, cnt1);
    k_scan<<<1, 512, 0, stream>>>(idx1, idx2, cnt1, loc1, loc2);

    const size_t total = (size_t)out_size;            // 1 + 2*S*E*C floats
    const size_t n4 = total >> 2;
    const size_t tail_start = n4 << 2;
    const size_t n_tail = total - tail_start;
    k_zero<<<4096, 256, 0, stream>>>((v4f*)out, n4, out, tail_start, n_tail);

    k_scatter<<<S_TOK / 256, 256, 0, stream>>>(idx1, idx2, loc1, loc2, g1, g2, gsum, cnt1, out);
}